// YOLOV3BBox_74156905332942
// MI455X (gfx1250) — compile-verified
//
#include <hip/hip_runtime.h>
#include <hip/hip_bf16.h>
#include <math.h>

// ---------------- problem constants ----------------
#define NCLS   80
#define NANCH  22743          // 3*(19^2 + 38^2 + 76^2)
#define NLOC   7581           // 19^2 + 38^2 + 76^2
#define TOPK   4096u
#define CAP    4608           // candidate cap (TOPK + tie slack)
#define MAXDET 100
#define BINS1  2048           // coarse hist: float bits >> 19
#define BINS2  4096           // refine hist: (bits >> 7) & 0xFFF
#define CONF_THR  0.005f
#define SCORE_THR 0.05f
#define IOU_THR   0.45f

// anchor base sizes (w,h) per level {stride32, stride16, stride8}
__constant__ float c_aw[3][3] = {{116.f,156.f,373.f},{30.f,62.f,59.f},{10.f,16.f,33.f}};
__constant__ float c_ah[3][3] = {{90.f,198.f,326.f},{61.f,45.f,119.f},{13.f,30.f,23.f}};

__device__ __forceinline__ float sigf(float x){ return 1.f/(1.f + __expf(-x)); }

// ---------------- CDNA5 async global->LDS (ASYNCcnt path) ----------------
__device__ __forceinline__ unsigned lds_addr_of(const void* p){
  // generic pointers to LDS: addr[31:0] is the wave-relative LDS byte address
  return (unsigned)(unsigned long long)p;
}
__device__ __forceinline__ void async_ld_b32(unsigned lds, const void* g){
  asm volatile("global_load_async_to_lds_b32 %0, %1, off"
               :: "v"(lds), "v"((unsigned long long)g) : "memory");
}
__device__ __forceinline__ void async_ld_b128(unsigned lds, const void* g){
  asm volatile("global_load_async_to_lds_b128 %0, %1, off"
               :: "v"(lds), "v"((unsigned long long)g) : "memory");
}
__device__ __forceinline__ void wait_async0(){
  asm volatile("s_wait_asynccnt 0" ::: "memory");
}

// ---------------- misc ----------------
__global__ void zero_u32(unsigned* __restrict__ p, int n){
  int i = blockIdx.x*256 + threadIdx.x;
  if (i < n) p[i] = 0u;
}

// ---------------- decode / histogram / compact (templated pass) ----------------
// MODE 0: write boxes + coarse histogram
// MODE 1: refine histogram within coarse boundary bin
// MODE 2: compact candidates with key >= threshold
template<int MODE>
__global__ void yolo_pass(const float* __restrict__ p0, const float* __restrict__ p1,
                          const float* __restrict__ p2,
                          float* __restrict__ boxes,          // [B,NANCH,4]
                          unsigned* __restrict__ hist1,       // [B,BINS1]
                          unsigned* __restrict__ hist2,       // [B,BINS2]
                          const unsigned* __restrict__ meta,  // [B,4]: T, A, found, thr
                          unsigned* __restrict__ cnt,         // [B]
                          float* __restrict__ cscore,         // [B,CAP]
                          unsigned* __restrict__ cidx,        // [B,CAP]
                          int* __restrict__ clabel)           // [B,CAP]
{
  __shared__ unsigned lh[BINS2];
  const int b   = blockIdx.y;
  const int tid = threadIdx.x;
  const int BINS = (MODE == 0) ? BINS1 : BINS2;

  if (MODE < 2){
    for (int i = tid; i < BINS; i += 256) lh[i] = 0u;
    __syncthreads();
  }

  unsigned Tm = 0u, thr = 0u;
  if (MODE == 1) Tm  = meta[b*4 + 0];
  if (MODE == 2) thr = meta[b*4 + 3];

  const int loc = blockIdx.x*256 + tid;
  if (loc < NLOC){
    int level, W, HW, locbase; float s;
    if (loc < 361)       { level=0; W=19; HW=361;  locbase=0;    s=32.f; }
    else if (loc < 1805) { level=1; W=38; HW=1444; locbase=361;  s=16.f; }
    else                 { level=2; W=76; HW=5776; locbase=1805; s=8.f;  }
    const int li = loc - locbase;
    const float* pm = (level==0) ? p0 : ((level==1) ? p1 : p2);
    const float fx = (float)(li % W), fy = (float)(li / W);

    for (int a = 0; a < 3; ++a){
      const float* base = pm + (size_t)(b*255 + a*85) * HW + li;
      const float obj = sigf(base[(size_t)4*HW]);

      if (MODE == 0){
        const float tx = base[0], ty = base[(size_t)HW];
        const float tw = base[(size_t)2*HW], th = base[(size_t)3*HW];
        const float cx = (fx + 0.5f)*s + (sigf(tx) - 0.5f)*s;
        const float cy = (fy + 0.5f)*s + (sigf(ty) - 0.5f)*s;
        const float hw = 0.5f * c_aw[level][a] * __expf(tw);
        const float hh = 0.5f * c_ah[level][a] * __expf(th);
        float4 bx; bx.x = cx - hw; bx.y = cy - hh; bx.z = cx + hw; bx.w = cy + hh;
        ((float4*)boxes)[(size_t)b*NANCH + (size_t)loc*3 + a] = bx;
      }

      const bool objok = (obj >= CONF_THR);
      const float* cp = base + (size_t)5*HW;
      for (int c = 0; c < NCLS; ++c){
        const float sc = sigf(cp[(size_t)c*HW]);
        if (objok && sc > SCORE_THR){
          const float score = sc * obj;              // in (0,1): positive float bits are order-preserving
          const unsigned key = __float_as_uint(score);
          if (MODE == 0){
            atomicAdd(&lh[key >> 19], 1u);
          } else if (MODE == 1){
            if ((key >> 19) == Tm) atomicAdd(&lh[(key >> 7) & 0xFFFu], 1u);
          } else {
            if (key >= thr){
              const unsigned pos = atomicAdd(&cnt[b], 1u);
              if (pos < CAP){
                const size_t o = (size_t)b*CAP + pos;
                cscore[o] = score;
                cidx[o]   = (unsigned)(loc*3 + a);
                clabel[o] = c;
              }
            }
          }
        }
      }
    }
  }

  if (MODE < 2){
    __syncthreads();
    unsigned* gh = (MODE == 0) ? (hist1 + (size_t)b*BINS1) : (hist2 + (size_t)b*BINS2);
    for (int i = tid; i < BINS; i += 256)
      if (lh[i]) atomicAdd(&gh[i], lh[i]);
  }
}

// find coarse boundary bin per image
__global__ void scan1(const unsigned* __restrict__ hist1, unsigned* __restrict__ meta){
  const int b = blockIdx.x;
  if (threadIdx.x != 0) return;
  unsigned cum = 0, A = 0, found = 0; int T = 0;
  for (int bin = BINS1 - 1; bin >= 0; --bin){
    const unsigned h = hist1[(size_t)b*BINS1 + bin];
    if (cum + h >= TOPK){ T = bin; A = cum; found = 1u; break; }
    cum += h;
  }
  meta[b*4+0] = (unsigned)T; meta[b*4+1] = A; meta[b*4+2] = found;
}

// refine within boundary bin -> final 25-bit threshold key
__global__ void scan2(const unsigned* __restrict__ hist2, unsigned* __restrict__ meta){
  const int b = blockIdx.x;
  if (threadIdx.x != 0) return;
  const unsigned T = meta[b*4+0], A = meta[b*4+1], found = meta[b*4+2];
  unsigned thr = 0u;
  if (found){
    unsigned cum = A; int R = 0;
    for (int r = BINS2 - 1; r >= 0; --r){
      const unsigned h = hist2[(size_t)b*BINS2 + r];
      if (cum + h >= TOPK){ R = r; break; }
      cum += h;
    }
    thr = (T << 19) | ((unsigned)R << 7);
  }
  meta[b*4+3] = thr;
}

// ---------------- NMS: one workgroup per image, whole working set in LDS ----------------
// dynamic LDS layout: float4 sbx[CAP] | float ssc[CAP] | int slb[CAP] | float soff[80]
//                     | float rv[1024] | int ri[1024] | float bc[8]
#define NMS_SMEM ((size_t)CAP*16 + (size_t)CAP*4 + (size_t)CAP*4 + 80*4 + 1024*4 + 1024*4 + 8*4)

__global__ void nms_kernel(const float* __restrict__ boxes,
                           const unsigned* __restrict__ cnt,
                           const float* __restrict__ cscore,
                           const unsigned* __restrict__ cidx,
                           const int* __restrict__ clabel,
                           float* __restrict__ dout, int B)
{
  extern __shared__ unsigned char smem[];
  float4* sbx  = (float4*)smem;
  float*  ssc  = (float*)(sbx + CAP);
  int*    slb  = (int*)(ssc + CAP);
  float*  soff = (float*)(slb + CAP);
  float*  rv   = soff + NCLS;
  int*    ri   = (int*)(rv + 1024);
  float*  bc   = (float*)(ri + 1024);

  const int b   = blockIdx.x;
  const int tid = threadIdx.x;
  unsigned K = cnt[b]; if (K > CAP) K = CAP;

  // async-gather candidates (score, label, box4) straight into LDS
  for (unsigned i = tid; i < K; i += 1024){
    const unsigned n = cidx[(size_t)b*CAP + i];
    const float4* gb = (const float4*)boxes + (size_t)b*NANCH + n;
    async_ld_b128(lds_addr_of(&sbx[i]), gb);
    async_ld_b32 (lds_addr_of(&ssc[i]), &cscore[(size_t)b*CAP + i]);
    async_ld_b32 (lds_addr_of(&slb[i]), &clabel[(size_t)b*CAP + i]);
  }
  wait_async0();
  __syncthreads();

  // per-class offset trick: off[c] = c * (max_coord + 1)
  float m = 0.f;
  for (unsigned i = tid; i < K; i += 1024){
    const float4 v = sbx[i];
    m = fmaxf(m, fmaxf(fmaxf(v.x, v.y), fmaxf(v.z, v.w)));
  }
  rv[tid] = m; __syncthreads();
  for (int o = 512; o > 0; o >>= 1){
    if (tid < o) rv[tid] = fmaxf(rv[tid], rv[tid + o]);
    __syncthreads();
  }
  const float maxc = rv[0];
  __syncthreads();
  if (tid < NCLS) soff[tid] = (float)tid * (maxc + 1.f);
  __syncthreads();

  float* dets = dout + (size_t)b*MAXDET*5;
  int*   labs = (int*)(dout + (size_t)B*MAXDET*5) + (size_t)b*MAXDET;
  const float NEGINF = -__builtin_inff();

  for (int it = 0; it < MAXDET; ++it){
    // block-wide argmax over alive scores
    float bv = NEGINF; int bi = -1;
    for (unsigned i = tid; i < K; i += 1024){
      const float s = ssc[i];
      if (s > bv){ bv = s; bi = (int)i; }
    }
    rv[tid] = bv; ri[tid] = bi; __syncthreads();
    for (int o = 512; o > 0; o >>= 1){
      if (tid < o && rv[tid + o] > rv[tid]){ rv[tid] = rv[tid + o]; ri[tid] = ri[tid + o]; }
      __syncthreads();
    }
    if (tid == 0){
      const float s = rv[0]; const int j = ri[0];
      const bool keep = (j >= 0) && (s > 0.f);
      float4 bx; bx.x = bx.y = bx.z = bx.w = 0.f;
      float off = 0.f; int lb = -1;
      if (keep){ bx = sbx[j]; lb = slb[j]; off = soff[lb]; }
      dets[it*5+0] = bx.x; dets[it*5+1] = bx.y;
      dets[it*5+2] = bx.z; dets[it*5+3] = bx.w;
      dets[it*5+4] = keep ? s : 0.f;
      labs[it] = keep ? lb : -1;
      if (j >= 0) ssc[j] = NEGINF;
      bc[0] = keep ? 1.f : 0.f;
      bc[1] = bx.x + off; bc[2] = bx.y + off; bc[3] = bx.z + off; bc[4] = bx.w + off;
    }
    __syncthreads();
    if (bc[0] > 0.f){
      const float wx1 = bc[1], wy1 = bc[2], wx2 = bc[3], wy2 = bc[4];
      const float wa = (wx2 - wx1) * (wy2 - wy1);
      for (unsigned i = tid; i < K; i += 1024){
        const float4 v = sbx[i];
        const float o = soff[slb[i]];
        const float x1 = v.x + o, y1 = v.y + o, x2 = v.z + o, y2 = v.w + o;
        float iw = fminf(wx2, x2) - fmaxf(wx1, x1);
        float ih = fminf(wy2, y2) - fmaxf(wy1, y1);
        iw = fmaxf(iw, 0.f); ih = fmaxf(ih, 0.f);
        const float inter = iw * ih;
        const float a2 = (x2 - x1) * (y2 - y1);
        const float iou = inter / (wa + a2 - inter + 1e-12f);
        if (iou > IOU_THR) ssc[i] = NEGINF;
      }
    }
    __syncthreads();
  }
}

// ---------------- host launch ----------------
extern "C" void kernel_launch(void* const* d_in, const int* in_sizes, int n_in,
                              void* d_out, int out_size, void* d_ws, size_t ws_size,
                              hipStream_t stream) {
  const float* p0 = (const float*)d_in[0];   // [B,255,19,19]
  const float* p1 = (const float*)d_in[1];   // [B,255,38,38]
  const float* p2 = (const float*)d_in[2];   // [B,255,76,76]
  const int B = in_sizes[0] / (255 * 361);

  // workspace layout
  char* w = (char*)d_ws;
  float* boxes = (float*)w;                 w += (size_t)B * NANCH * 4 * sizeof(float); // float4-aligned
  float*    cscore = (float*)w;             w += (size_t)B * CAP * sizeof(float);
  unsigned* cidx   = (unsigned*)w;          w += (size_t)B * CAP * sizeof(unsigned);
  int*      clabel = (int*)w;               w += (size_t)B * CAP * sizeof(int);
  unsigned* hist1  = (unsigned*)w;          w += (size_t)B * BINS1 * sizeof(unsigned);
  unsigned* hist2  = (unsigned*)w;          w += (size_t)B * BINS2 * sizeof(unsigned);
  unsigned* meta   = (unsigned*)w;          w += (size_t)B * 4 * sizeof(unsigned);
  unsigned* cnt    = (unsigned*)w;          w += (size_t)B * sizeof(unsigned);

  // 1) zero hist1 | hist2 | meta | cnt (contiguous)
  const int nz = B * (BINS1 + BINS2 + 4 + 1);
  zero_u32<<<(nz + 255) / 256, 256, 0, stream>>>(hist1, nz);

  const dim3 pgrid((NLOC + 255) / 256, B);
  // 2) decode boxes + coarse histogram
  yolo_pass<0><<<pgrid, 256, 0, stream>>>(p0,p1,p2, boxes, hist1, hist2, meta, cnt, cscore, cidx, clabel);
  // 3) coarse threshold
  scan1<<<B, 32, 0, stream>>>(hist1, meta);
  // 4) refine histogram
  yolo_pass<1><<<pgrid, 256, 0, stream>>>(p0,p1,p2, boxes, hist1, hist2, meta, cnt, cscore, cidx, clabel);
  // 5) final threshold key
  scan2<<<B, 32, 0, stream>>>(hist2, meta);
  // 6) compact top candidates
  yolo_pass<2><<<pgrid, 256, 0, stream>>>(p0,p1,p2, boxes, hist1, hist2, meta, cnt, cscore, cidx, clabel);

  // 7) NMS, one WGP per image, working set resident in the 320KB LDS
  hipFuncSetAttribute((const void*)nms_kernel,
                      hipFuncAttributeMaxDynamicSharedMemorySize, (int)NMS_SMEM);
  nms_kernel<<<B, 1024, NMS_SMEM, stream>>>(boxes, cnt, cscore, cidx, clabel, (float*)d_out, B);
}